// P2MLoss_44521630991154
// MI455X (gfx1250) — compile-verified
//
#include <hip/hip_runtime.h>
#include <hip/hip_bf16.h>

typedef __attribute__((ext_vector_type(2))) float v2f;
typedef __attribute__((ext_vector_type(8))) float v8f;

#define INF_BITS 0x7F800000u
#define TILES_PER_CHUNK 128

// ---------------------------------------------------------------------------
// Workspace layout (32-bit words):
//   [0..11]          : accumulators: d1sum[3], d2sum[3], edgesum[3], lapsum[3]
//   [64 .. 64+8191]  : d1min uint buffer (per-gt-point min), capacity 8192
//   [8256 .. ]       : d2min uint buffer (per-pred-point min), capacity 40962
// ---------------------------------------------------------------------------

__global__ void zero_acc_kernel(float* __restrict__ acc) {
    if (threadIdx.x < 12) acc[threadIdx.x] = 0.0f;
}

__global__ void fill_inf_kernel(unsigned* __restrict__ buf, int n) {
    int i = blockIdx.x * blockDim.x + threadIdx.x;
    int stride = gridDim.x * blockDim.x;
    for (; i < n; i += stride) buf[i] = INF_BITS;
}

// ---------------------------------------------------------------------------
// One wave per (16-row gt tile) x (chunk of pred tiles).
// Distance tile via V_WMMA_F32_16X16X4_F32:
//   A[m]   = (gx, gy, gz, 1)
//   B[:,n] = (-2bx, -2by, -2bz, |b|^2)
//   tile   = -2 a.b + |b|^2 ;  d = max(tile + |a|^2, 0)
// Row mins (over pred) -> atomicMin into d1min[gt index]
// Col mins (over gt)   -> atomicMin into d2min[pred index]
// Both are monotone uint-min on non-negative floats: fully deterministic.
// ---------------------------------------------------------------------------
__device__ __forceinline__ v2f make_bfrag(const float* __restrict__ pred,
                                          int p, int np, int hi) {
    float bx = 0.0f, by = 0.0f, bz = 0.0f, bn2 = 1.0e30f;  // pad -> huge dist
    if (p < np) {
        bx  = pred[p * 3 + 0];
        by  = pred[p * 3 + 1];
        bz  = pred[p * 3 + 2];
        bn2 = bx * bx + by * by + bz * bz;
    }
    v2f b;
    b.x = hi ? (-2.0f * bz) : (-2.0f * bx);
    b.y = hi ? bn2 : (-2.0f * by);
    return b;
}

__global__ __launch_bounds__(32)
void chamfer_tile_kernel(const float* __restrict__ gt, int ng,
                         const float* __restrict__ pred, int np,
                         unsigned* __restrict__ d1min,
                         unsigned* __restrict__ d2min) {
    const int lane = threadIdx.x;          // 0..31 (wave32)
    const int n    = lane & 15;            // N column within tile
    const int hi   = lane >> 4;            // which K/M half this lane serves
    const int gbase = blockIdx.x * 16;

    // ---- A fragment: lane holds A[m][2*hi], A[m][2*hi+1], m = lane&15
    int m = gbase + n;
    if (m >= ng) m = ng - 1;               // safety (ng is a multiple of 16)
    const float gx = gt[m * 3 + 0];
    const float gy = gt[m * 3 + 1];
    const float gz = gt[m * 3 + 2];
    v2f a;
    a.x = hi ? gz : gx;
    a.y = hi ? 1.0f : gy;

    // ---- per-lane |a|^2 for the 8 rows this lane's accumulator holds:
    // VGPR r <-> row M = r + 8*hi
    float anv[8];
#pragma unroll
    for (int r = 0; r < 8; ++r) {
        int mm = gbase + r + 8 * hi;
        if (mm >= ng) mm = ng - 1;
        const float x = gt[mm * 3 + 0];
        const float y = gt[mm * 3 + 1];
        const float z = gt[mm * 3 + 2];
        anv[r] = x * x + y * y + z * z;
    }

    float rowmin[8];
#pragma unroll
    for (int r = 0; r < 8; ++r) rowmin[r] = 3.4e38f;

    const int ptiles = (np + 15) >> 4;
    const int t0 = blockIdx.y * TILES_PER_CHUNK;
    int t1 = t0 + TILES_PER_CHUNK;
    if (t1 > ptiles) t1 = ptiles;

    // Unroll-by-2: two independent WMMAs in flight per iteration.
    for (int t = t0; t < t1; t += 2) {
        // prefetch 2 tiles ahead (lowers to global_prefetch_b8 on gfx1250)
        {
            const int pf = (t + 2) * 16 + n;
            if (pf < np) __builtin_prefetch(&pred[pf * 3], 0, 1);
        }

        const int pA = t * 16 + n;
        const v2f b0 = make_bfrag(pred, pA, np, hi);
        v8f c0 = {};
        c0 = __builtin_amdgcn_wmma_f32_16x16x4_f32(
                false, a, false, b0, (short)0, c0, false, false);

        const bool have2 = (t + 1) < t1;
        const int pB = (t + 1) * 16 + n;
        v8f c1 = {};
        if (have2) {
            const v2f b1 = make_bfrag(pred, pB, np, hi);
            c1 = __builtin_amdgcn_wmma_f32_16x16x4_f32(
                    false, a, false, b1, (short)0, c1, false, false);
        }

        float cmin0 = 3.4e38f, cmin1 = 3.4e38f;
#pragma unroll
        for (int r = 0; r < 8; ++r) {
            float d0 = fmaxf(c0[r] + anv[r], 0.0f);
            rowmin[r] = fminf(rowmin[r], d0);
            cmin0 = fminf(cmin0, d0);
        }
        if (have2) {
#pragma unroll
            for (int r = 0; r < 8; ++r) {
                float d1 = fmaxf(c1[r] + anv[r], 0.0f);
                rowmin[r] = fminf(rowmin[r], d1);
                cmin1 = fminf(cmin1, d1);
            }
        }
        // fold the two M-halves (lanes L and L^16 hold the same column N)
        cmin0 = fminf(cmin0, __shfl_xor(cmin0, 16, 32));
        cmin1 = fminf(cmin1, __shfl_xor(cmin1, 16, 32));
        if (hi == 0) {
            if (pA < np) atomicMin(&d2min[pA], __float_as_uint(cmin0));
            if (have2 && pB < np) atomicMin(&d2min[pB], __float_as_uint(cmin1));
        }
    }

    // reduce rowmin across the 16 lanes of each half (tile columns)
#pragma unroll
    for (int r = 0; r < 8; ++r) {
        float v = rowmin[r];
        v = fminf(v, __shfl_xor(v, 1, 32));
        v = fminf(v, __shfl_xor(v, 2, 32));
        v = fminf(v, __shfl_xor(v, 4, 32));
        v = fminf(v, __shfl_xor(v, 8, 32));
        rowmin[r] = v;
    }
    if (n == 0) {  // lanes 0 (rows 0..7) and 16 (rows 8..15)
#pragma unroll
        for (int r = 0; r < 8; ++r) {
            const int row = gbase + r + 8 * hi;
            if (row < ng)
                atomicMin(&d1min[row], __float_as_uint(rowmin[r]));
        }
    }
}

// Single-block deterministic sum of a uint-encoded float min buffer.
__global__ __launch_bounds__(256)
void reduce_minbuf_kernel(const unsigned* __restrict__ buf, int n,
                          float* __restrict__ out) {
    __shared__ float smem[256];
    float s = 0.0f;
    for (int i = threadIdx.x; i < n; i += 256) s += __uint_as_float(buf[i]);
    smem[threadIdx.x] = s;
    __syncthreads();
    for (int off = 128; off > 0; off >>= 1) {
        if ((int)threadIdx.x < off) smem[threadIdx.x] += smem[threadIdx.x + off];
        __syncthreads();
    }
    if (threadIdx.x == 0) *out = smem[0];
}

__global__ __launch_bounds__(256)
void edge_kernel(const float* __restrict__ pred,
                 const int* __restrict__ edges, int ne,
                 float* __restrict__ out) {
    __shared__ float smem[256];
    float s = 0.0f;
    for (int i = threadIdx.x; i < ne; i += 256) {
        const int e0 = edges[2 * i + 0];
        const int e1 = edges[2 * i + 1];
        const float dx = pred[3 * e0 + 0] - pred[3 * e1 + 0];
        const float dy = pred[3 * e0 + 1] - pred[3 * e1 + 1];
        const float dz = pred[3 * e0 + 2] - pred[3 * e1 + 2];
        s += dx * dx + dy * dy + dz * dz;
    }
    smem[threadIdx.x] = s;
    __syncthreads();
    for (int off = 128; off > 0; off >>= 1) {
        if ((int)threadIdx.x < off) smem[threadIdx.x] += smem[threadIdx.x + off];
        __syncthreads();
    }
    if (threadIdx.x == 0) *out = smem[0];
}

__global__ __launch_bounds__(256)
void lap_kernel(const float* __restrict__ feats,
                const float* __restrict__ pts,
                const int* __restrict__ lap, int nv,
                float* __restrict__ out) {
    __shared__ float smem[256];
    float s = 0.0f;
    for (int v = threadIdx.x; v < nv; v += 256) {
        float sfx = 0.f, sfy = 0.f, sfz = 0.f;
        float spx = 0.f, spy = 0.f, spz = 0.f;
#pragma unroll
        for (int j = 0; j < 8; ++j) {
            const int idx = lap[10 * v + j];
            if (idx >= 0) {
                sfx += feats[3 * idx + 0];
                sfy += feats[3 * idx + 1];
                sfz += feats[3 * idx + 2];
                spx += pts[3 * idx + 0];
                spy += pts[3 * idx + 1];
                spz += pts[3 * idx + 2];
            }
        }
        const float inv_cnt = 1.0f / (float)lap[10 * v + 9];
        const float l1x = feats[3 * v + 0] - sfx * inv_cnt;
        const float l1y = feats[3 * v + 1] - sfy * inv_cnt;
        const float l1z = feats[3 * v + 2] - sfz * inv_cnt;
        const float l2x = pts[3 * v + 0] - spx * inv_cnt;
        const float l2y = pts[3 * v + 1] - spy * inv_cnt;
        const float l2z = pts[3 * v + 2] - spz * inv_cnt;
        const float dx = l1x - l2x, dy = l1y - l2y, dz = l1z - l2z;
        s += dx * dx + dy * dy + dz * dz;
    }
    smem[threadIdx.x] = s;
    __syncthreads();
    for (int off = 128; off > 0; off >>= 1) {
        if ((int)threadIdx.x < off) smem[threadIdx.x] += smem[threadIdx.x + off];
        __syncthreads();
    }
    if (threadIdx.x == 0) *out = smem[0];
}

__global__ void finalize_kernel(const float* __restrict__ acc,
                                float* __restrict__ out, int ng,
                                int v0, int v1, int v2,
                                int e0, int e1, int e2) {
    if (threadIdx.x != 0 || blockIdx.x != 0) return;
    const float invng = 1.0f / (float)ng;
    const float chamfer = acc[0] * invng + acc[3] / (float)v0 +
                          acc[1] * invng + acc[4] / (float)v1 +
                          acc[2] * invng + acc[5] / (float)v2;
    // mean(diff^2 over (E,3)) * 3 == sumsq / E
    const float edge = acc[6] / (float)e0 + acc[7] / (float)e1 +
                       acc[8] / (float)e2;
    // mean(d^2 over (V,3)) * 3 == sumsq / V ; lap_const = {0.2, 1.0, 1.0}
    const float lap = 0.2f * acc[9] / (float)v0 + acc[10] / (float)v1 +
                      acc[11] / (float)v2;
    out[0] = 100.0f * chamfer + 0.1f * edge + 0.3f * lap;
    out[1] = chamfer;
    out[2] = edge;
    out[3] = lap;
}

extern "C" void kernel_launch(void* const* d_in, const int* in_sizes, int n_in,
                              void* d_out, int out_size, void* d_ws,
                              size_t ws_size, hipStream_t stream) {
    // Input order: per level l: pred_pts, pred_feats, edges, lap_idx; then gt.
    const float* pred_pts[3]   = {(const float*)d_in[0], (const float*)d_in[4],
                                  (const float*)d_in[8]};
    const float* pred_feats[3] = {(const float*)d_in[1], (const float*)d_in[5],
                                  (const float*)d_in[9]};
    const int* edges[3]   = {(const int*)d_in[2], (const int*)d_in[6],
                             (const int*)d_in[10]};
    const int* lap_idx[3] = {(const int*)d_in[3], (const int*)d_in[7],
                             (const int*)d_in[11]};
    const float* gt = (const float*)d_in[12];

    int V[3], E[3];
    for (int l = 0; l < 3; ++l) {
        V[l] = in_sizes[4 * l] / 3;
        E[l] = in_sizes[4 * l + 2] / 2;
    }
    const int ng = in_sizes[12] / 3;

    float* acc = (float*)d_ws;                      // 12 accumulators
    unsigned* d1min = (unsigned*)d_ws + 64;         // per-gt min  (<= 8192)
    unsigned* d2min = (unsigned*)d_ws + 64 + 8192;  // per-pred min (<= 40962)
    float* out = (float*)d_out;

    zero_acc_kernel<<<1, 32, 0, stream>>>(acc);

    const int gtiles = (ng + 15) / 16;
    for (int l = 0; l < 3; ++l) {
        const int nv = V[l];
        const int ptiles = (nv + 15) / 16;
        const int pchunks = (ptiles + TILES_PER_CHUNK - 1) / TILES_PER_CHUNK;

        fill_inf_kernel<<<(ng + 255) / 256, 256, 0, stream>>>(d1min, ng);
        fill_inf_kernel<<<(nv + 255) / 256, 256, 0, stream>>>(d2min, nv);

        dim3 grid(gtiles, pchunks, 1);
        chamfer_tile_kernel<<<grid, 32, 0, stream>>>(gt, ng, pred_pts[l], nv,
                                                     d1min, d2min);

        reduce_minbuf_kernel<<<1, 256, 0, stream>>>(d1min, ng, &acc[0] + l);
        reduce_minbuf_kernel<<<1, 256, 0, stream>>>(d2min, nv, &acc[3] + l);

        edge_kernel<<<1, 256, 0, stream>>>(pred_pts[l], edges[l], E[l],
                                           &acc[6] + l);
        lap_kernel<<<1, 256, 0, stream>>>(pred_feats[l], pred_pts[l],
                                          lap_idx[l], nv, &acc[9] + l);
    }

    finalize_kernel<<<1, 32, 0, stream>>>(acc, out, ng, V[0], V[1], V[2], E[0],
                                          E[1], E[2]);
}